// DiceLoss_79310866088659
// MI455X (gfx1250) — compile-verified
//
#include <hip/hip_runtime.h>

// ---------------------------------------------------------------------------
// Dice loss, B=8 C=19 H=W=512, float32 logits, int32 targets, IGNORE=255.
// Bandwidth-bound streaming reduction:  ~167 MB single pass -> ~7 us floor.
// ---------------------------------------------------------------------------

#define NCLS 19
#define IGNORE_IDX 255

constexpr int kB   = 8;
constexpr int kHW  = 512 * 512;          // 262144
constexpr int kBHW = kB * kHW;           // 2097152 pixels
constexpr int kBlock = 256;              // 8 waves of 32
constexpr int kGroups = 4;               // groups of 4 pixels per thread
// pixels per block = 256 * 4 * 4 = 4096 ; grid = 2097152 / 4096 = 512

typedef float v2f __attribute__((ext_vector_type(2)));
typedef float v8f __attribute__((ext_vector_type(8)));

// ---------------------------------------------------------------------------
// Zero the tiny workspace accumulators (ws is poisoned by the harness).
// ---------------------------------------------------------------------------
__global__ void dice_zero_ws(float* __restrict__ wsf, unsigned* __restrict__ wsc) {
    int i = threadIdx.x;
    if (i < 2 * NCLS) wsf[i] = 0.0f;       // probs_sum[19], inter[19]
    if (i < NCLS)     wsc[i] = 0u;         // counts[19]
}

// ---------------------------------------------------------------------------
// Main streaming kernel: softmax over C per pixel + per-class reductions.
// ---------------------------------------------------------------------------
__global__ void __launch_bounds__(kBlock)
dice_main(const float* __restrict__ logits,
          const int*   __restrict__ targets,
          float*       __restrict__ g_psum,    // [19]
          float*       __restrict__ g_inter,   // [19]
          unsigned*    __restrict__ g_cnt)     // [19]
{
    __shared__ float    s_psum[NCLS];
    __shared__ float    s_inter[NCLS];
    __shared__ unsigned s_cnt[NCLS];

    const int tid = threadIdx.x;
    if (tid < NCLS) { s_psum[tid] = 0.0f; s_inter[tid] = 0.0f; s_cnt[tid] = 0u; }
    __syncthreads();

    float acc[NCLS];                       // per-lane probs_sum partials
#pragma unroll
    for (int c = 0; c < NCLS; ++c) acc[c] = 0.0f;

    const int blockBase = blockIdx.x * (kBlock * kGroups * 4);  // block never
    const int b  = blockBase / kHW;        // crosses a batch boundary (4096|HW)
    const float* __restrict__ bbase = logits + b * (NCLS * kHW);

    for (int g = 0; g < kGroups; ++g) {
        const int p0 = blockBase + g * (kBlock * 4) + tid * 4;  // 4 consecutive px
        const int hw = p0 - b * kHW;
        const float* __restrict__ src = bbase + hw;

        float4 v[NCLS];
#pragma unroll
        for (int c = 0; c < NCLS; ++c)
            v[c] = *(const float4*)(src + c * kHW);             // global_load_b128
        const int4 t4 = *(const int4*)(targets + p0);

        // channel-wise max (per pixel component)
        float4 m = v[0];
#pragma unroll
        for (int c = 1; c < NCLS; ++c) {
            m.x = fmaxf(m.x, v[c].x); m.y = fmaxf(m.y, v[c].y);
            m.z = fmaxf(m.z, v[c].z); m.w = fmaxf(m.w, v[c].w);
        }

        float4 z = make_float4(0.f, 0.f, 0.f, 0.f);
        float4 gsel = make_float4(0.f, 0.f, 0.f, 0.f);   // exp at target class
#pragma unroll
        for (int c = 0; c < NCLS; ++c) {
            float4 e;
            e.x = __expf(v[c].x - m.x); e.y = __expf(v[c].y - m.y);
            e.z = __expf(v[c].z - m.z); e.w = __expf(v[c].w - m.w);
            z.x += e.x; z.y += e.y; z.z += e.z; z.w += e.w;
            v[c] = e;                                    // keep exp in regs
            gsel.x = (t4.x == c) ? e.x : gsel.x;
            gsel.y = (t4.y == c) ? e.y : gsel.y;
            gsel.z = (t4.z == c) ? e.z : gsel.z;
            gsel.w = (t4.w == c) ? e.w : gsel.w;
        }

        const float rx = 1.0f / z.x, ry = 1.0f / z.y;
        const float rz = 1.0f / z.z, rw = 1.0f / z.w;
        const bool vx = (t4.x != IGNORE_IDX), vy = (t4.y != IGNORE_IDX);
        const bool vz = (t4.z != IGNORE_IDX), vw = (t4.w != IGNORE_IDX);
        // fold validity mask into the softmax normalizer: invalid -> weight 0
        const float fx = vx ? rx : 0.0f, fy = vy ? ry : 0.0f;
        const float fz = vz ? rz : 0.0f, fw = vw ? rw : 0.0f;

#pragma unroll
        for (int c = 0; c < NCLS; ++c)
            acc[c] += v[c].x * fx + v[c].y * fy + v[c].z * fz + v[c].w * fw;

        if (vx) { atomicAdd(&s_inter[t4.x], gsel.x * rx); atomicAdd(&s_cnt[t4.x], 1u); }
        if (vy) { atomicAdd(&s_inter[t4.y], gsel.y * ry); atomicAdd(&s_cnt[t4.y], 1u); }
        if (vz) { atomicAdd(&s_inter[t4.z], gsel.z * rz); atomicAdd(&s_cnt[t4.z], 1u); }
        if (vw) { atomicAdd(&s_inter[t4.w], gsel.w * rw); atomicAdd(&s_cnt[t4.w], 1u); }
    }

    // wave32 butterfly reduction of probs_sum partials, then one LDS add/wave
#pragma unroll
    for (int c = 0; c < NCLS; ++c) {
        float a = acc[c];
        a += __shfl_xor(a, 16, 32);
        a += __shfl_xor(a,  8, 32);
        a += __shfl_xor(a,  4, 32);
        a += __shfl_xor(a,  2, 32);
        a += __shfl_xor(a,  1, 32);
        if ((tid & 31) == 0) atomicAdd(&s_psum[c], a);
    }
    __syncthreads();

    if (tid < NCLS) {
        atomicAdd(&g_psum[tid],  s_psum[tid]);
        atomicAdd(&g_inter[tid], s_inter[tid]);
        atomicAdd(&g_cnt[tid],   s_cnt[tid]);
    }
}

// ---------------------------------------------------------------------------
// Finalize: one wave32. dice per class, then 32-lane sum of (1-dice) done with
// V_WMMA_F32_16X16X4_F32 against an all-ones B matrix.
//   A layout (16x4 f32): lane L<16 -> (M=L, K=0/1); lane L>=16 -> (M=L-16, K=2/3)
//   With A = {val, 0}, B = ones:  D[m][n] = val[m] + val[m+16]  for all n.
//   Summing a lane's 8 D-VGPRs: lanes 0-15 -> sum_{m=0..7}, lanes 16-31 -> m=8..15.
// ---------------------------------------------------------------------------
__global__ void dice_finalize(const float*    __restrict__ g_psum,
                              const float*    __restrict__ g_inter,
                              const unsigned* __restrict__ g_cnt,
                              float*          __restrict__ out)
{
    const int lane = threadIdx.x;                 // 32 threads, EXEC all ones
    const int idx  = (lane < NCLS) ? lane : 0;    // avoid OOB, stays branchless

    const float ps = g_psum[idx];
    const float it = g_inter[idx];
    const float cf = (float)g_cnt[idx];

    const float uni  = ps + cf;
    const float dice = (2.0f * it + 1.0f) / (uni + 1.0f);
    float val  = (lane < NCLS) ? (1.0f - dice) : 0.0f;
    float cval = (lane < NCLS) ? cf : 0.0f;

    // total valid-pixel count (scalar) via shuffle butterfly
    cval += __shfl_xor(cval, 16, 32);
    cval += __shfl_xor(cval,  8, 32);
    cval += __shfl_xor(cval,  4, 32);
    cval += __shfl_xor(cval,  2, 32);
    cval += __shfl_xor(cval,  1, 32);

    // 32-lane sum of val via WMMA f32 16x16x4 with B = ones
    v2f A;  A[0] = val;  A[1] = 0.0f;
    v2f Bm; Bm[0] = 1.0f; Bm[1] = 1.0f;
    v8f Cm = {};
    v8f D = __builtin_amdgcn_wmma_f32_16x16x4_f32(
        /*neg_a=*/false, A, /*neg_b=*/false, Bm,
        /*c_mod=*/(short)0, Cm, /*reuse_a=*/false, /*reuse_b=*/false);

    float part = D[0] + D[1] + D[2] + D[3] + D[4] + D[5] + D[6] + D[7];
    // lane 0 holds sum over m=0..7 of (val[m]+val[m+16]); lane 16 holds m=8..15
    float total = __shfl(part, 0, 32) + __shfl(part, 16, 32);

    if (lane == 0) {
        float loss = total * (1.0f / (float)NCLS);
        out[0] = (cval > 0.0f) ? loss : 0.0f;     // no-valid-pixels fallback
    }
}

// ---------------------------------------------------------------------------
extern "C" void kernel_launch(void* const* d_in, const int* in_sizes, int n_in,
                              void* d_out, int out_size, void* d_ws, size_t ws_size,
                              hipStream_t stream)
{
    const float* logits  = (const float*)d_in[0];
    const int*   targets = (const int*)d_in[1];

    float*    wsf    = (float*)d_ws;
    float*    psum   = wsf;             // [19]
    float*    inter  = wsf + NCLS;      // [19]
    unsigned* cnt    = (unsigned*)(wsf + 2 * NCLS);  // [19]

    dice_zero_ws<<<1, 64, 0, stream>>>(wsf, cnt);

    const int grid = kBHW / (kBlock * kGroups * 4);  // 512 blocks
    dice_main<<<grid, kBlock, 0, stream>>>(logits, targets, psum, inter, cnt);

    dice_finalize<<<1, 32, 0, stream>>>(psum, inter, cnt, (float*)d_out);
}